// MultiMarginRankingLoss_79388175499583
// MI455X (gfx1250) — compile-verified
//
#include <hip/hip_runtime.h>
#include <hip/hip_bf16.h>
#include <math.h>

// MI455X / gfx1250, wave32. Pairwise rank term reduced with
// V_WMMA_F32_16X16X4_F32 (A = masked relu-diff tile, B = all-ones).

typedef __attribute__((ext_vector_type(2))) float v2f;
typedef __attribute__((ext_vector_type(8))) float v8f;

#define NQ 128
#define NP 1024
#define BLK 256   // 8 wave32 per block

__global__ __launch_bounds__(BLK)
void MultiMarginRankingLoss_rowkernel(const float* __restrict__ pred,
                                      const int*   __restrict__ y,
                                      float*       __restrict__ ws) {
    __shared__ float sx[NP];     // row's pred
    __shared__ float syf[NP];    // relevance mask (y==1) as float
    __shared__ float red_r[BLK];
    __shared__ float red_b[BLK];

    const int q   = blockIdx.x;
    const int tid = threadIdx.x;
    const float* __restrict__ xrow = pred + (size_t)q * NP;
    const int*   __restrict__ yrow = y    + (size_t)q * NP;

    // Stage row into LDS + fused BCE partial:
    //   bce_e = max(x,0) - x*yf + log1p(exp(-|x|))
    float bce = 0.0f;
    for (int p = tid; p < NP; p += BLK) {
        float x  = xrow[p];
        float yf = (yrow[p] == 1) ? 1.0f : 0.0f;
        sx[p]  = x;
        syf[p] = yf;
        bce += fmaxf(x, 0.0f) - x * yf + log1pf(expf(-fabsf(x)));
    }
    __syncthreads();

    const int wave = tid >> 5;   // 0..7
    const int lane = tid & 31;
    const int m    = lane & 15;  // A-matrix row M (both lane halves)
    const int hi   = lane >> 4;  // selects K pair {2hi, 2hi+1}

    const v2f ones = {1.0f, 1.0f};   // B = 4x16 all-ones (layout-independent)

    // Each wave owns p-tiles wave, wave+8, ... (uniform: 64/8 = 8 tiles/wave,
    // so EXEC is all-ones at every WMMA).
    float rank_acc = 0.0f;   // wave-aggregate counts every D entry -> 16x true sum
    for (int pt = wave; pt < NP / 16; pt += 8) {
        const int   p   = pt * 16 + m;
        const float xp  = sx[p];
        const float nrp = 1.0f - syf[p];    // non-relevant mask for row p

        v8f acc = {};   // 16x16 f32 accumulator, starts at 0
        for (int jt = 0; jt < NP / 16; ++jt) {
            const int jb = jt * 16 + 2 * hi;
#pragma unroll
            for (int ch = 0; ch < 4; ++ch) {
                const int j0 = jb + ch * 4;
                // A[m][k] = relu(x_p - x_j) * nonrel_p * rel_j for this lane's
                // two K slots. Masks folded into A; B stays all-ones.
                v2f a;
                a.x = fmaxf(xp - sx[j0],     0.0f) * (nrp * syf[j0]);
                a.y = fmaxf(xp - sx[j0 + 1], 0.0f) * (nrp * syf[j0 + 1]);
                acc = __builtin_amdgcn_wmma_f32_16x16x4_f32(
                        /*neg_a=*/false, a,
                        /*neg_b=*/false, ones,
                        /*c_mod=*/(short)0, acc,
                        /*reuse_a=*/false, /*reuse_b=*/false);
            }
        }
        // Sum this lane's 8 D entries; summing over the whole wave later
        // counts each of the 256 tile entries exactly once (= 16x row total).
        rank_acc += acc[0] + acc[1] + acc[2] + acc[3]
                  + acc[4] + acc[5] + acc[6] + acc[7];
    }

    red_r[tid] = rank_acc;
    red_b[tid] = bce;
    __syncthreads();

    if (tid == 0) {
        float sr = 0.0f, sb = 0.0f;
        for (int i = 0; i < BLK; ++i) { sr += red_r[i]; sb += red_b[i]; }
        // sr = 16 * S_q (column replication); store B_q - S_q for row q.
        ws[q] = sb - sr * (1.0f / 16.0f);
    }
}

__global__ void MultiMarginRankingLoss_final(const float* __restrict__ ws,
                                             float* __restrict__ out) {
    if (threadIdx.x == 0 && blockIdx.x == 0) {
        // Deterministic serial fold of 128 per-row partials.
        float s = 0.0f;
        for (int i = 0; i < NQ; ++i) s += ws[i];
        // out = sum_q [ -0.5*S_q + 0.5*B_q ] / (128*1024)
        out[0] = s * (0.5f / (128.0f * 1024.0f));
    }
}

extern "C" void kernel_launch(void* const* d_in, const int* in_sizes, int n_in,
                              void* d_out, int out_size, void* d_ws, size_t ws_size,
                              hipStream_t stream) {
    const float* pred = (const float*)d_in[0];   // [128,1024] f32
    const int*   yv   = (const int*)d_in[1];     // [128,1024] i32 (0/1)
    float* out = (float*)d_out;                  // scalar f32
    float* ws  = (float*)d_ws;                   // >= 128 floats of scratch

    MultiMarginRankingLoss_rowkernel<<<NQ, BLK, 0, stream>>>(pred, yv, ws);
    MultiMarginRankingLoss_final<<<1, 32, 0, stream>>>(ws, out);
}